// TrajectoryInformedSplatAttention_43224550867601
// MI455X (gfx1250) — compile-verified
//
#include <hip/hip_runtime.h>
#include <hip/hip_bf16.h>
#include <math.h>

typedef float v2f __attribute__((ext_vector_type(2)));
typedef float v8f __attribute__((ext_vector_type(8)));

#define BB   4
#define SS   4096
#define DD   1024
#define MAXD 8
#define SM1  (SS - 1)

// ---------------------------------------------------------------------------
// Pass 1: per (b,t) edge, compute mag = ||e[t+1]-e[t]||, store
//   magw[t] = tanh(2*mag), rmag[t] = (mag > 1e-6) ? 1/mag : 0
// One 256-thread block per edge; D=1024 -> exactly one float4 per thread.
// ---------------------------------------------------------------------------
__global__ __launch_bounds__(256) void tisa_mag_kernel(const float* __restrict__ emb,
                                                       float* __restrict__ magw,
                                                       float* __restrict__ rmag) {
    __shared__ float red[256];
    const int row = blockIdx.x;            // b*(S-1) + t
    const int b   = row / SM1;
    const int t   = row - b * SM1;
    const float* e0 = emb + ((size_t)b * SS + t) * DD;
    const float* e1 = e0 + DD;
    const int tid = threadIdx.x;

    float4 x0 = *(const float4*)(e0 + tid * 4);
    float4 x1 = *(const float4*)(e1 + tid * 4);
    float dx = x1.x - x0.x, dy = x1.y - x0.y, dz = x1.z - x0.z, dw = x1.w - x0.w;
    red[tid] = dx * dx + dy * dy + dz * dz + dw * dw;
    __syncthreads();
#pragma unroll
    for (int o = 128; o > 0; o >>= 1) {
        if (tid < o) red[tid] += red[tid + o];
        __syncthreads();
    }
    if (tid == 0) {
        float mag  = sqrtf(red[0]);
        magw[row]  = tanhf(2.0f * mag);
        rmag[row]  = (mag > 1e-6f) ? (1.0f / mag) : 0.0f;
    }
}

// ---------------------------------------------------------------------------
// Pass 2: traj = A_banded @ E via V_WMMA_F32_16X16X4_F32 (full fp32).
// Block: 16 positions x 256 channels, 128 threads = 4 wave32.
//   A[m][r] = c(p, g-1) - c(p, g), g = p0-8+r,  r = 0..23
//   c(p,t)  = lin(p, j=t-p+8) * magw[t] * rmag[t] / wsum(p)
// ---------------------------------------------------------------------------
#define ROWS  24           // 16 positions + MAXD window rows
#define LSTR  260          // LDS row stride (floats), avoids bank conflicts
#define ASTR  25

__global__ __launch_bounds__(128) void tisa_traj_kernel(const float* __restrict__ emb,
                                                        const float* __restrict__ magw,
                                                        const float* __restrict__ rmag,
                                                        float* __restrict__ out) {
    __shared__ float Eb[ROWS * LSTR];      // 24 x 256 embedding tile (padded)
    __shared__ float As[16 * ASTR];        // 16 x 24 banded coefficient tile

    const int p0  = blockIdx.x * 16;       // position tile
    const int c0  = blockIdx.y * 256;      // channel chunk
    const int b   = blockIdx.z;
    const int tid = threadIdx.x;

    const float* mwB = magw + (size_t)b * SM1;
    const float* rmB = rmag + (size_t)b * SM1;
    const float* eB  = emb + (size_t)b * SS * DD;

    // ---- stage embedding rows p0-8 .. p0+15 (cols c0..c0+255) into LDS ----
    for (int i = tid; i < ROWS * 64; i += 128) {
        int r  = i >> 6;                   // 0..23
        int c4 = (i & 63) << 2;            // 0..252 step 4
        int g  = p0 - MAXD + r;
        float4 v = make_float4(0.f, 0.f, 0.f, 0.f);
        if (g >= 0) v = *(const float4*)(eB + (size_t)g * DD + c0 + c4);
        float* dst = &Eb[r * LSTR + c4];
        dst[0] = v.x; dst[1] = v.y; dst[2] = v.z; dst[3] = v.w;
    }

    // ---- build banded A coefficients (telescoped weights) in LDS ----
    for (int e = tid; e < 16 * ROWS; e += 128) {
        int m = e / ROWS, r = e - m * ROWS;
        int p = p0 + m;
        int wsize   = (p < MAXD) ? p : MAXD;
        float denom = (float)((wsize - 1 > 1) ? (wsize - 1) : 1);
        float wsum  = 0.f;
#pragma unroll
        for (int j = 0; j < MAXD; ++j) {
            int t = p - MAXD + j;
            if (t < 0) continue;
            float k   = (float)(j - (MAXD - wsize));
            float lin = (wsize > 1) ? (0.1f + 0.9f * k / denom) : 0.1f;
            wsum += lin * mwB[t];
        }
        wsum = fmaxf(wsum, 1e-8f);
        float inv = 1.0f / wsum;
        auto cval = [&](int j) -> float {
            if (j < 0 || j >= MAXD) return 0.f;
            int t = p - MAXD + j;
            if (t < 0) return 0.f;
            float k   = (float)(j - (MAXD - wsize));
            float lin = (wsize > 1) ? (0.1f + 0.9f * k / denom) : 0.1f;
            return lin * mwB[t] * rmB[t] * inv;
        };
        int j2 = r - m;                    // -c at row g
        As[m * ASTR + r] = cval(j2 - 1) - cval(j2);
    }
    __syncthreads();

    const int lane = tid & 31;
    const int wave = tid >> 5;
    const int lg   = lane >> 4;            // lane group 0/1
    const int ln   = lane & 15;

    // A fragments (shared across all channel tiles of this wave):
    // lane m (group g), VGPR v  <->  A[m][4*ks + 2*g + v]
    v2f afr[6];
#pragma unroll
    for (int ks = 0; ks < 6; ++ks) {
        int k0 = ks * 4 + 2 * lg;
        afr[ks].x = As[ln * ASTR + k0];
        afr[ks].y = As[ln * ASTR + k0 + 1];
    }

    // Each wave handles 4 of the 16 channel tiles.
#pragma unroll
    for (int tt = 0; tt < 4; ++tt) {
        int ct = wave * 4 + tt;            // 0..15
        int cc = ct * 16 + ln;             // column within the 256-chunk
        v8f acc = {};
#pragma unroll
        for (int ks = 0; ks < 6; ++ks) {
            int k0 = ks * 4 + 2 * lg;
            v2f bfr;                       // B[k][n]: n = lane%16, k = 2*(lane/16)+v
            bfr.x = Eb[k0 * LSTR + cc];
            bfr.y = Eb[(k0 + 1) * LSTR + cc];
            acc = __builtin_amdgcn_wmma_f32_16x16x4_f32(
                false, afr[ks], false, bfr, (short)0, acc, false, false);
        }
        // D[m][n]: VGPR v -> row p0 + v + 8*lg, col = c0 + ct*16 + ln
        float* op = out + ((size_t)b * SS + p0 + 8 * lg) * DD + c0 + ct * 16 + ln;
#pragma unroll
        for (int v = 0; v < 8; ++v) {
            op[(size_t)v * DD] = acc[v];
        }
    }
}

// ---------------------------------------------------------------------------
extern "C" void kernel_launch(void* const* d_in, const int* in_sizes, int n_in,
                              void* d_out, int out_size, void* d_ws, size_t ws_size,
                              hipStream_t stream) {
    const float* emb = (const float*)d_in[0];
    float* out  = (float*)d_out;
    float* magw = (float*)d_ws;                       // B*(S-1) floats
    float* rmag = magw + (size_t)BB * SM1;            // B*(S-1) floats

    // Pass 1: edge magnitudes -> magw / rmag
    tisa_mag_kernel<<<dim3(BB * SM1), dim3(256), 0, stream>>>(emb, magw, rmag);

    // Pass 2: banded WMMA GEMM -> traj
    dim3 grid(SS / 16, DD / 256, BB);
    tisa_traj_kernel<<<grid, dim3(128), 0, stream>>>(emb, magw, rmag, out);
}